// ArrayMultiplier24x24_43860206027313
// MI455X (gfx1250) — compile-verified
//
#include <hip/hip_runtime.h>
#include <hip/hip_bf16.h>
#include <stdint.h>

// ---------------------------------------------------------------------------
// 24x24 -> 48-bit binary multiplier on float bit-vectors.
// Per element: pack 24 {0.0,1.0} floats into a uint32 for each operand,
// one 32x32->64 integer multiply, unpack 48 product bits to floats.
// Data movement: TDM (tensor_load_to_lds) DMAs each workgroup's 24KB input
// slabs global->LDS (TENSORcnt), compute reads from LDS, results stream out
// with float4 (B128) stores. Memory-bound: ~50MB @ 23.3TB/s ~= 2.2us floor.
// ---------------------------------------------------------------------------

typedef unsigned int u32x4 __attribute__((ext_vector_type(4)));
typedef int          i32x4 __attribute__((ext_vector_type(4)));
typedef int          i32x8 __attribute__((ext_vector_type(8)));

#define ELEMS_PER_BLOCK 256
#define NBITS 24
#define OBITS 48

// 1-D contiguous tile load, global -> LDS, via the Tensor Data Mover.
// ndwords 4-byte elements starting at gaddr land at lds_byte_addr.
__device__ __forceinline__ void tdm_load_1d_to_lds(uint32_t lds_byte_addr,
                                                   uint64_t gaddr,
                                                   uint32_t ndwords) {
  // D# group 0 (128b): count=1 (valid, user mode), lds_addr @ [63:32],
  // global_addr[56:0] @ [120:64], type=2 @ [127:126].
  u32x4 g0 = { 1u,
               lds_byte_addr,
               (uint32_t)gaddr,
               ((uint32_t)((gaddr >> 32) & 0x01FFFFFFull)) | 0x80000000u };
  // D# group 1 (256b): workgroup_mask=0 (not in cluster), data_size=2 (4B),
  // no pad/iterate/atomic-barrier. tensor_dim0 @ [79:48], tile_dim0 @ [127:112],
  // tensor_dim0_stride @ [207:160]. tile_dim1/2 = 0 (unused -> 1-D tile).
  i32x8 g1 = { (int)0x00020000,                    // data_size=2, everything else 0
               (int)((ndwords & 0xFFFFu) << 16),   // tensor_dim0[15:0] -> bits 63:48
               (int)((ndwords >> 16) & 0xFFFFu),   // tensor_dim0[31:16]; tensor_dim1=0
               (int)((ndwords & 0xFFFFu) << 16),   // tensor_dim1 hi=0; tile_dim0 -> 127:112
               0,                                  // tile_dim1=0, tile_dim2=0
               (int)ndwords,                       // tensor_dim0_stride[31:0]
               0, 0 };
  i32x4 gz4 = { 0, 0, 0, 0 };                      // groups 2/3 unused (<=2-D tensor)
  i32x8 gz8 = { 0, 0, 0, 0, 0, 0, 0, 0 };         // extra group (6-arg toolchain form)
  __builtin_amdgcn_tensor_load_to_lds(g0, g1, gz4, gz4, gz8, 0);
}

__device__ __forceinline__ uint32_t f01_bit(float x) {
  // x is exactly 0.0f (0x00000000) or 1.0f (0x3F800000): exponent bit 23.
  return (__float_as_uint(x) >> 23) & 1u;
}

__global__ __launch_bounds__(ELEMS_PER_BLOCK)
void ArrayMultiplier24x24_kernel(const float* __restrict__ A,
                                 const float* __restrict__ B,
                                 float* __restrict__ out,
                                 int n) {
  __shared__ __align__(16) float lA[ELEMS_PER_BLOCK * NBITS];
  __shared__ __align__(16) float lB[ELEMS_PER_BLOCK * NBITS];

  const int     tid  = threadIdx.x;
  const int64_t base = (int64_t)blockIdx.x * ELEMS_PER_BLOCK;
  const int     cnt  = min(ELEMS_PER_BLOCK, n - (int)base);

  // One wave issues two TDM DMAs (A-slab, B-slab) and waits on TENSORcnt;
  // the workgroup barrier releases everyone else once LDS is populated.
  if (tid == 0) {
    const uint32_t ndw = (uint32_t)cnt * (uint32_t)NBITS;   // <= 6144, fits tile_dim0
    tdm_load_1d_to_lds((uint32_t)(uintptr_t)&lA[0],
                       (uint64_t)(uintptr_t)(A + base * NBITS), ndw);
    tdm_load_1d_to_lds((uint32_t)(uintptr_t)&lB[0],
                       (uint64_t)(uintptr_t)(B + base * NBITS), ndw);
    __builtin_amdgcn_s_wait_tensorcnt(0);
  }
  __syncthreads();

  if (tid < cnt) {
    // Pack LSB-first bit vectors into integers.
    uint32_t ai = 0u, bi = 0u;
#pragma unroll
    for (int j = 0; j < NBITS; ++j) {
      ai |= f01_bit(lA[tid * NBITS + j]) << j;
      bi |= f01_bit(lB[tid * NBITS + j]) << j;
    }

    // Exact 24x24 -> 48-bit product.
    const uint64_t p = (uint64_t)ai * (uint64_t)bi;

    // Unpack 48 bits to floats, 16B-vectorized stores (192B per thread,
    // contiguous & 16B-aligned since 48 floats = 12 float4).
    float4* o4 = reinterpret_cast<float4*>(out + (base + tid) * (int64_t)OBITS);
#pragma unroll
    for (int q = 0; q < OBITS / 4; ++q) {
      float4 v;
      v.x = (float)((uint32_t)(p >> (4 * q + 0)) & 1u);
      v.y = (float)((uint32_t)(p >> (4 * q + 1)) & 1u);
      v.z = (float)((uint32_t)(p >> (4 * q + 2)) & 1u);
      v.w = (float)((uint32_t)(p >> (4 * q + 3)) & 1u);
      o4[q] = v;
    }
  }
}

extern "C" void kernel_launch(void* const* d_in, const int* in_sizes, int n_in,
                              void* d_out, int out_size, void* d_ws, size_t ws_size,
                              hipStream_t stream) {
  const float* A   = (const float*)d_in[0];
  const float* B   = (const float*)d_in[1];
  float*       out = (float*)d_out;
  const int n      = in_sizes[0] / NBITS;            // batch elements (131072)
  const int blocks = (n + ELEMS_PER_BLOCK - 1) / ELEMS_PER_BLOCK;
  ArrayMultiplier24x24_kernel<<<blocks, ELEMS_PER_BLOCK, 0, stream>>>(A, B, out, n);
}